// FuzzyCMeansLoss_24189255811658
// MI455X (gfx1250) — compile-verified
//
#include <hip/hip_runtime.h>

typedef __attribute__((ext_vector_type(2))) float v2f;
typedef __attribute__((ext_vector_type(8))) float v8f;

#define GWG   128       // phase-1 workgroups (fixed -> deterministic partials)
#define BATCH 100000
#define KDIM  32        // clusters
#define FDIM  128       // features
#define EPSF  1e-8f

// ---------------------------------------------------------------------------
// Phase 1: per-WG partial num = Wm^T X  (32x128) via V_WMMA_F32_16X16X4_F32,
// per-half colsum partials, and partial sum of x2[b]*sw[b].
// 512 threads = 16 wave32 waves; wave (mtile,ntile) owns one 16x16 num tile.
// Batch loop unrolled 4x (16 rows/iter) so loads use immediate offsets.
// ---------------------------------------------------------------------------
__global__ __launch_bounds__(512) void fcm_phase1(const float* __restrict__ X,
                                                  const float* __restrict__ W,
                                                  float* __restrict__ ws) {
  const int g      = blockIdx.x;
  const int tid    = threadIdx.x;
  const int wave   = tid >> 5;
  const int lane   = tid & 31;
  const int lane16 = lane & 15;
  const int half   = lane >> 4;     // which 16-lane half of the wave
  const int mtile  = wave >> 3;     // 0..1  (cluster tile)
  const int ntile  = wave & 7;      // 0..7  (feature tile)

  const int chunk = (BATCH + GWG - 1) / GWG;
  const int start = g * chunk;
  const int end   = (start + chunk < BATCH) ? (start + chunk) : BATCH;

  float* pnum = ws;                                   // [GWG][32][128]
  float* pcol = ws + (size_t)GWG * (KDIM * FDIM);     // [GWG][2][32]
  float* px2  = pcol + (size_t)GWG * (2 * KDIM);      // [GWG]

  const int col  = mtile * 16 + lane16;   // cluster index 0..31
  const int feat = ntile * 16 + lane16;   // feature index 0..127

  v8f acc = {};
  float csum = 0.0f;

  // Per-lane base pointers: fp32 A 16x4 / B 4x16 layout puts this lane at
  // batch row  b0 + 2*half + v,  so bake (start + 2*half) into the base and
  // index steps with compile-time immediate offsets.
  const float* wBase = W + (size_t)(start + half * 2) * KDIM + col;
  const float* xBase = X + (size_t)(start + half * 2) * FDIM + feat;

  int b0 = start;
  // ---- main body: 16 batch rows / iteration, 4 WMMAs, no bounds checks ----
  for (; b0 + 16 <= end; b0 += 16) {
    __builtin_prefetch(xBase + 16 * FDIM, 0, 1);   // global_prefetch_b8
    __builtin_prefetch(wBase + 16 * KDIM, 0, 1);
#pragma unroll
    for (int i = 0; i < 4; ++i) {
      v2f a, bx;
      float w0 = wBase[(i * 4 + 0) * KDIM];
      float w1 = wBase[(i * 4 + 1) * KDIM];
      a[0] = w0 * w0;                       // Wm = W^2
      a[1] = w1 * w1;
      bx[0] = xBase[(i * 4 + 0) * FDIM];
      bx[1] = xBase[(i * 4 + 1) * FDIM];
      csum += a[0] + a[1];                  // per-lane colsum partial
      // D(16x16 f32) = A(16x4) * B(4x16) + C -> v_wmma_f32_16x16x4_f32
      acc = __builtin_amdgcn_wmma_f32_16x16x4_f32(false, a, false, bx,
                                                  (short)0, acc, false, false);
    }
    wBase += 16 * KDIM;
    xBase += 16 * FDIM;
  }
  // ---- predicated tail: clamp addresses, zero A; no EXEC branch near WMMA --
  for (; b0 < end; b0 += 4) {
    v2f a, bx;
#pragma unroll
    for (int v = 0; v < 2; ++v) {
      int b  = b0 + half * 2 + v;
      int bc = (b < end) ? b : 0;
      float w = W[(size_t)bc * KDIM + col];
      w = (b < end) ? (w * w) : 0.0f;
      a[v]  = w;
      bx[v] = X[(size_t)bc * FDIM + feat];
    }
    csum += a[0] + a[1];
    acc = __builtin_amdgcn_wmma_f32_16x16x4_f32(false, a, false, bx,
                                                (short)0, acc, false, false);
  }

  // Write this wave's 16x16 tile of the partial num matrix.
  // C/D layout: VGPR r -> row = 8*half + r, col = lane16 (within tile).
  float* tileOut = pnum + (size_t)g * (KDIM * FDIM);
#pragma unroll
  for (int r = 0; r < 8; ++r) {
    int row = mtile * 16 + half * 8 + r;
    tileOut[row * FDIM + feat] = acc[r];
  }

  // colsum partials: the two ntile==0 waves cover all 32 clusters; each lane
  // half saw a disjoint batch subset, so store per-half and sum in phase 2.
  if (ntile == 0) {
    pcol[(size_t)g * (2 * KDIM) + half * KDIM + col] = csum;
  }

  // ---- second (L2-resident) pass: partial  sum_b x2[b] * sw[b] ----
  float local = 0.0f;
  for (int b = start + tid; b < end; b += 512) {
    const float4* xr = (const float4*)(X + (size_t)b * FDIM);
    float x2 = 0.0f;
#pragma unroll
    for (int i = 0; i < FDIM / 4; ++i) {
      float4 v = xr[i];
      x2 += v.x * v.x + v.y * v.y + v.z * v.z + v.w * v.w;
    }
    const float4* wr = (const float4*)(W + (size_t)b * KDIM);
    float sw = 0.0f;
#pragma unroll
    for (int i = 0; i < KDIM / 4; ++i) {
      float4 v = wr[i];
      sw += v.x * v.x + v.y * v.y + v.z * v.z + v.w * v.w;   // sum of Wm row
    }
    local += x2 * sw;
  }
  __shared__ float red[512];
  red[tid] = local;
  __syncthreads();
  for (int s = 256; s > 0; s >>= 1) {
    if (tid < s) red[tid] += red[tid + s];
    __syncthreads();
  }
  if (tid == 0) px2[g] = red[0];
}

// ---------------------------------------------------------------------------
// Phase 2: single WG. Reduce the GWG partial slots, form centroids, and emit
//   loss = ( x2sw - 2*sum(C .* num) + sum_k colsum[k]*||c_k||^2 ) / (B*K)
// ---------------------------------------------------------------------------
__global__ __launch_bounds__(128) void fcm_phase2(const float* __restrict__ ws,
                                                  float* __restrict__ out) {
  __shared__ float numS[KDIM * FDIM];
  __shared__ float colS[KDIM];
  __shared__ float x2swS;
  __shared__ float r2[128], r3[128];

  const float* pnum = ws;
  const float* pcol = ws + (size_t)GWG * (KDIM * FDIM);
  const float* px2  = pcol + (size_t)GWG * (2 * KDIM);

  const int tid = threadIdx.x;   // 0..127 == feature index

  for (int idx = tid; idx < KDIM * FDIM; idx += 128) {
    float s = 0.0f;
    for (int g = 0; g < GWG; ++g) s += pnum[(size_t)g * (KDIM * FDIM) + idx];
    numS[idx] = s;
  }
  if (tid < KDIM) {
    float s = 0.0f;
    for (int g = 0; g < GWG; ++g) {
      s += pcol[(size_t)g * (2 * KDIM) + tid];
      s += pcol[(size_t)g * (2 * KDIM) + KDIM + tid];
    }
    colS[tid] = s;
  }
  if (tid == 0) {
    float s = 0.0f;
    for (int g = 0; g < GWG; ++g) s += px2[g];
    x2swS = s;
  }
  __syncthreads();

  float t2 = 0.0f, t3 = 0.0f;
  for (int k = 0; k < KDIM; ++k) {
    float n = numS[k * FDIM + tid];
    float c = n / (colS[k] + EPSF);   // centroid element
    t2 += c * n;                      // sum C .* num
    t3 += colS[k] * c * c;            // sum_k colsum[k] * ||c_k||^2
  }
  r2[tid] = t2;
  r3[tid] = t3;
  __syncthreads();
  for (int s = 64; s > 0; s >>= 1) {
    if (tid < s) { r2[tid] += r2[tid + s]; r3[tid] += r3[tid + s]; }
    __syncthreads();
  }
  if (tid == 0) {
    out[0] = (x2swS - 2.0f * r2[0] + r3[0]) /
             ((float)BATCH * (float)KDIM);
  }
}

extern "C" void kernel_launch(void* const* d_in, const int* in_sizes, int n_in,
                              void* d_out, int out_size, void* d_ws, size_t ws_size,
                              hipStream_t stream) {
  const float* X = (const float*)d_in[0];   // (100000, 128) fp32
  const float* W = (const float*)d_in[1];   // (100000, 32)  fp32
  float* out = (float*)d_out;               // scalar fp32 loss
  float* ws  = (float*)d_ws;                // needs GWG*(4096+64+1)*4 ~= 2.1 MB

  fcm_phase1<<<GWG, 512, 0, stream>>>(X, W, ws);
  fcm_phase2<<<1, 128, 0, stream>>>(ws, out);
}